// TransConv_11974368821797
// MI455X (gfx1250) — compile-verified
//
#include <hip/hip_runtime.h>

// ---------------------------------------------------------------------------
// MI455X (gfx1250) implementation.
//
// ~172 GFLOP fp32; GEMM working sets (h=32MB, W=1MB, kvs=16MB) are L2-resident
// (192MB L2), so the problem is fp32-matrix-compute bound -> use
// V_WMMA_F32_16X16X4_F32 (full precision, 2048 FLOP/instr).
//
// All GEMMs share one scheme: block = 4 waves, block tile 128(M) x 64(N),
// K panel of 32 staged in LDS.  A is stored row-major As[m][k]; B is stored
// TRANSPOSED Bst[n][k] so both A and B fragments are contiguous 8-byte LDS
// loads (ds_load_2addr_b64-mergeable), no cross-lane repacking.
// Each wave computes 2 m-subtiles x 4 n-subtiles = 8 WMMA per k-step with
// only 6 LDS loads.
//
// WMMA fp32 lane layouts (ISA 7.12.2):
//   A (16x4):  lane<16: M=lane,    v0=A[M][k0+0], v1=A[M][k0+1]
//              lane>=16:M=lane-16, v0=A[M][k0+2], v1=A[M][k0+3]
//   B (4x16):  lane<16: N=lane,    v0=B[k0+0][N], v1=B[k0+1][N]
//              lane>=16:N=lane-16, v0=B[k0+2][N], v1=B[k0+3][N]
//   C/D (16x16): VGPR r: lanes0-15 -> (M=r,   N=lane),
//                        lanes16-31-> (M=r+8, N=lane-16)
// ---------------------------------------------------------------------------

typedef float v2f __attribute__((ext_vector_type(2)));
typedef float v8f __attribute__((ext_vector_type(8)));

#define TILE_K 32
#define LPAD 4           // row stride TILE_K+4 = 36 floats = 144B (16B aligned)
#define BM 128
#define BN 64

__device__ __forceinline__ v8f v8f_zero() {
  v8f z;
#pragma unroll
  for (int i = 0; i < 8; ++i) z[i] = 0.0f;
  return z;
}

__device__ __forceinline__ v8f wmma_f32(v2f a, v2f b, v8f c) {
  return __builtin_amdgcn_wmma_f32_16x16x4_f32(false, a, false, b, (short)0, c,
                                               false, false);
}

__device__ __forceinline__ float block_sum_256(float v, float* red) {
  const int t = threadIdx.x;
  red[t] = v;
  __syncthreads();
#pragma unroll
  for (int s = 128; s > 0; s >>= 1) {
    if (t < s) red[t] += red[t + s];
    __syncthreads();
  }
  float r = red[0];
  __syncthreads();
  return r;
}

// ---------------------------------------------------------------------------
// Graph structure
// ---------------------------------------------------------------------------
__global__ void tc_zero_i(int* p, int n) {
  int i = blockIdx.x * blockDim.x + threadIdx.x;
  if (i < n) p[i] = 0;
}

__global__ void tc_count(const int* __restrict__ batch, int* __restrict__ counts,
                         int N) {
  int i = blockIdx.x * blockDim.x + threadIdx.x;
  if (i < N) atomicAdd(&counts[batch[i]], 1);
}

__global__ void tc_starts(const int* __restrict__ counts, int* __restrict__ starts,
                          int B) {
  if (threadIdx.x == 0 && blockIdx.x == 0) {
    int s = 0;
    for (int b = 0; b < B; ++b) { starts[b] = s; s += counts[b]; }
  }
}

// ---------------------------------------------------------------------------
// fc0 + LN(ln0) + ReLU   (one block of 256 per node; Cin=16, D=256)
// ---------------------------------------------------------------------------
__global__ __launch_bounds__(256) void tc_fc0_ln_relu(
    const float* __restrict__ x, const float* __restrict__ w,
    const float* __restrict__ bias, const float* __restrict__ lnw,
    const float* __restrict__ lnb, float* __restrict__ h, int Cin, int D) {
  __shared__ float xr[16];
  __shared__ float red[256];
  const int n = blockIdx.x;
  const int t = threadIdx.x;
  if (t < Cin) xr[t] = x[(size_t)n * Cin + t];
  __syncthreads();
  float acc = bias[t];
#pragma unroll
  for (int k = 0; k < 16; ++k) acc = fmaf(xr[k], w[k * D + t], acc);
  const float mean = block_sum_256(acc, red) * (1.0f / 256.0f);
  const float d = acc - mean;
  const float var = block_sum_256(d * d, red) * (1.0f / 256.0f);
  const float y = d * rsqrtf(var + 1e-5f) * lnw[t] + lnb[t];
  h[(size_t)n * D + t] = fmaxf(y, 0.0f);
}

// ---------------------------------------------------------------------------
// Shared WMMA inner product over one staged K panel.
// As[m][k] row-major, Bst[n][k] transposed; wave covers rows
// [wave*32, wave*32+32), all 64 columns; acc[st][nt].
// ---------------------------------------------------------------------------
__device__ __forceinline__ void wmma_panel(
    const float (*As)[TILE_K + LPAD], const float (*Bst)[TILE_K + LPAD],
    int wave, int half, int l16, v8f acc[2][4]) {
#pragma unroll
  for (int kk = 0; kk < TILE_K; kk += 4) {
    const int ka = kk + 2 * half;
    v2f a0 = *(const v2f*)&As[wave * 32 + l16][ka];
    v2f a1 = *(const v2f*)&As[wave * 32 + 16 + l16][ka];
#pragma unroll
    for (int nt = 0; nt < 4; ++nt) {
      v2f b = *(const v2f*)&Bst[nt * 16 + l16][ka];
      acc[0][nt] = wmma_f32(a0, b, acc[0][nt]);
      acc[1][nt] = wmma_f32(a1, b, acc[1][nt]);
    }
  }
}

// ---------------------------------------------------------------------------
// GEMM:  Out(MxNout) = A(MxK,row-major) @ W(KxNout,row-major) + bias
// M%128==0, K%32==0, Nout%64==0.
// ---------------------------------------------------------------------------
__global__ __launch_bounds__(128) void tc_gemm_hw(
    const float* __restrict__ A, const float* __restrict__ W,
    const float* __restrict__ bias, float* __restrict__ Out, int M, int K,
    int Nout) {
  __shared__ float As[BM][TILE_K + LPAD];
  __shared__ float Bst[BN][TILE_K + LPAD];
  const int t = threadIdx.x;
  const int wave = t >> 5, lane = t & 31, half = lane >> 4, l16 = lane & 15;
  const int m0 = blockIdx.y * BM;
  const int n0 = blockIdx.x * BN;

  v8f acc[2][4];
#pragma unroll
  for (int s = 0; s < 2; ++s)
#pragma unroll
    for (int i = 0; i < 4; ++i) acc[s][i] = v8f_zero();

  for (int k0 = 0; k0 < K; k0 += TILE_K) {
    if (k0 + TILE_K < K) {
      __builtin_prefetch(A + (size_t)(m0 + t) * K + k0 + TILE_K, 0, 0);
      __builtin_prefetch(W + (size_t)(k0 + TILE_K + (t & 31)) * Nout + n0, 0, 0);
    }
    __syncthreads();
    // A tile 128x32: row-major, vectorized 16B stores (rows are 16B aligned)
#pragma unroll
    for (int i = 0; i < 8; ++i) {
      int lin = t + i * 128;                    // 0..1023
      int row = lin >> 3, c4 = (lin & 7) * 4;   // 128 rows x 8 float4
      float4 va = *(const float4*)(A + (size_t)(m0 + row) * K + k0 + c4);
      *(float4*)&As[row][c4] = va;
    }
    // B tile 32x64 -> stored transposed Bst[n][k]
#pragma unroll
    for (int i = 0; i < 4; ++i) {
      int lin = t + i * 128;                    // 0..511
      int rb = lin >> 4, cb = (lin & 15) * 4;   // 32 rows x 16 float4
      float4 vb = *(const float4*)(W + (size_t)(k0 + rb) * Nout + n0 + cb);
      Bst[cb + 0][rb] = vb.x;
      Bst[cb + 1][rb] = vb.y;
      Bst[cb + 2][rb] = vb.z;
      Bst[cb + 3][rb] = vb.w;
    }
    __syncthreads();
    wmma_panel(As, Bst, wave, half, l16, acc);
  }
#pragma unroll
  for (int st = 0; st < 2; ++st)
#pragma unroll
    for (int nt = 0; nt < 4; ++nt) {
      const int col = n0 + nt * 16 + l16;
      const float bv = bias ? bias[col] : 0.0f;
#pragma unroll
      for (int r = 0; r < 8; ++r) {
        const int row = m0 + wave * 32 + st * 16 + half * 8 + r;
        Out[(size_t)row * Nout + col] = acc[st][nt][r] + bv;
      }
    }
}

// ---------------------------------------------------------------------------
// kvs[b,h](256x256) = sum over nodes of graph b: ks[n,h,:]^T vs[n,h,:]
// The transposed-LDS staging of ks provides the k^T for the A fragments.
// ---------------------------------------------------------------------------
__global__ __launch_bounds__(128) void tc_gemm_kvs(
    const float* __restrict__ Ks, const float* __restrict__ Vs,
    const int* __restrict__ starts, const int* __restrict__ counts,
    float* __restrict__ Kvs, int Hn, int Dh, int ldq) {
  __shared__ float Kat[BM][TILE_K + LPAD];   // Kat[m][k] = ks[node k][m]
  __shared__ float Vbt[BN][TILE_K + LPAD];   // Vbt[n][k] = vs[node k][n]
  const int t = threadIdx.x;
  const int wave = t >> 5, lane = t & 31, half = lane >> 4, l16 = lane & 15;
  const int bh = blockIdx.z;
  const int b = bh / Hn, h = bh % Hn;
  const int start = starts[b], count = counts[b];
  const int m0 = blockIdx.y * BM, n0 = blockIdx.x * BN;

  v8f acc[2][4];
#pragma unroll
  for (int s = 0; s < 2; ++s)
#pragma unroll
    for (int i = 0; i < 4; ++i) acc[s][i] = v8f_zero();

  for (int k0 = 0; k0 < count; k0 += TILE_K) {
    __syncthreads();
    // ks tile: 32 nodes x 128 m -> Kat[m][k]
#pragma unroll
    for (int i = 0; i < 8; ++i) {
      int lin = t + i * 128;                     // 0..1023
      int kr = lin >> 5, mc = (lin & 31) * 4;    // 32 rows x 32 float4
      float4 kv = {0.f, 0.f, 0.f, 0.f};
      if (k0 + kr < count)
        kv = *(const float4*)(Ks + (size_t)(start + k0 + kr) * ldq +
                              (size_t)h * Dh + m0 + mc);
      Kat[mc + 0][kr] = kv.x;
      Kat[mc + 1][kr] = kv.y;
      Kat[mc + 2][kr] = kv.z;
      Kat[mc + 3][kr] = kv.w;
    }
    // vs tile: 32 nodes x 64 d -> Vbt[n][k]
#pragma unroll
    for (int i = 0; i < 4; ++i) {
      int lin = t + i * 128;                     // 0..511
      int kr = lin >> 4, nc = (lin & 15) * 4;    // 32 rows x 16 float4
      float4 vv = {0.f, 0.f, 0.f, 0.f};
      if (k0 + kr < count)
        vv = *(const float4*)(Vs + (size_t)(start + k0 + kr) * ldq +
                              (size_t)h * Dh + n0 + nc);
      Vbt[nc + 0][kr] = vv.x;
      Vbt[nc + 1][kr] = vv.y;
      Vbt[nc + 2][kr] = vv.z;
      Vbt[nc + 3][kr] = vv.w;
    }
    __syncthreads();
    wmma_panel(Kat, Vbt, wave, half, l16, acc);
  }
  const size_t obase = (size_t)bh * Dh * Dh;
#pragma unroll
  for (int st = 0; st < 2; ++st)
#pragma unroll
    for (int nt = 0; nt < 4; ++nt) {
      const int col = n0 + nt * 16 + l16;
#pragma unroll
      for (int r = 0; r < 8; ++r) {
        const int row = m0 + wave * 32 + st * 16 + half * 8 + r;
        Kvs[obase + (size_t)row * Dh + col] = acc[st][nt][r];
      }
    }
}

// ---------------------------------------------------------------------------
// Numo[n,h,:] = qs[n,h,:] @ kvs[b,h]   (raw; scaling applied in epilogue)
// ---------------------------------------------------------------------------
__global__ __launch_bounds__(128) void tc_gemm_num(
    const float* __restrict__ Qs, const float* __restrict__ Kvs,
    const int* __restrict__ starts, const int* __restrict__ counts,
    float* __restrict__ Numo, int Hn, int Dh, int ldq, int tilesPerGraph) {
  __shared__ float As[BM][TILE_K + LPAD];
  __shared__ float Bst[BN][TILE_K + LPAD];
  const int t = threadIdx.x;
  const int wave = t >> 5, lane = t & 31, half = lane >> 4, l16 = lane & 15;
  const int b = blockIdx.y / tilesPerGraph;
  const int tb = blockIdx.y % tilesPerGraph;
  const int h = blockIdx.z;
  const int start = starts[b], count = counts[b];
  const int r0 = tb * BM;
  if (r0 >= count) return;
  const int n0 = blockIdx.x * BN;
  const size_t kvoff = (size_t)(b * Hn + h) * Dh * Dh;

  v8f acc[2][4];
#pragma unroll
  for (int s = 0; s < 2; ++s)
#pragma unroll
    for (int i = 0; i < 4; ++i) acc[s][i] = v8f_zero();

  for (int k0 = 0; k0 < Dh; k0 += TILE_K) {
    __syncthreads();
    // Q tile 128x32 (row-major)
#pragma unroll
    for (int i = 0; i < 8; ++i) {
      int lin = t + i * 128;
      int row = lin >> 3, c4 = (lin & 7) * 4;
      float4 va = {0.f, 0.f, 0.f, 0.f};
      if (r0 + row < count)
        va = *(const float4*)(Qs + (size_t)(start + r0 + row) * ldq +
                              (size_t)h * Dh + k0 + c4);
      *(float4*)&As[row][c4] = va;
    }
    // kvs tile 32x64 -> transposed
#pragma unroll
    for (int i = 0; i < 4; ++i) {
      int lin = t + i * 128;
      int rb = lin >> 4, cb = (lin & 15) * 4;
      float4 vb = *(const float4*)(Kvs + kvoff + (size_t)(k0 + rb) * Dh + n0 + cb);
      Bst[cb + 0][rb] = vb.x;
      Bst[cb + 1][rb] = vb.y;
      Bst[cb + 2][rb] = vb.z;
      Bst[cb + 3][rb] = vb.w;
    }
    __syncthreads();
    wmma_panel(As, Bst, wave, half, l16, acc);
  }
#pragma unroll
  for (int st = 0; st < 2; ++st)
#pragma unroll
    for (int nt = 0; nt < 4; ++nt) {
      const int col = n0 + nt * 16 + l16;
#pragma unroll
      for (int r = 0; r < 8; ++r) {
        const int rg = r0 + wave * 32 + st * 16 + half * 8 + r;
        if (rg < count)
          Numo[(size_t)(start + rg) * ldq + (size_t)h * Dh + col] =
              acc[st][nt][r];
      }
    }
}

// ---------------------------------------------------------------------------
// Deterministic reductions for ||q||^2, ||k||^2 and per-graph sum(k)
// ---------------------------------------------------------------------------
__global__ __launch_bounds__(256) void tc_sumsq_partial(
    const float* __restrict__ p, size_t n, float* __restrict__ part) {
  __shared__ float red[256];
  float local = 0.0f;
  for (size_t i = (size_t)blockIdx.x * blockDim.x + threadIdx.x; i < n;
       i += (size_t)gridDim.x * blockDim.x) {
    float v = p[i];
    local = fmaf(v, v, local);
  }
  float s = block_sum_256(local, red);
  if (threadIdx.x == 0) part[blockIdx.x] = s;
}

__global__ __launch_bounds__(256) void tc_sumsq_final(
    const float* __restrict__ part, float* __restrict__ out) {
  __shared__ float red[256];
  const int t = threadIdx.x;
  float local = part[t] + part[t + 256] + part[t + 512] + part[t + 768];
  float s = block_sum_256(local, red);
  if (t == 0) out[0] = s;
}

__global__ void tc_inv(float* scal) {
  if (threadIdx.x == 0 && blockIdx.x == 0)
    scal[2] = rsqrtf(scal[0] * scal[1]);  // 1 / (||q|| * ||k||)
}

__global__ __launch_bounds__(256) void tc_ksum_part(
    const float* __restrict__ Ks, const int* __restrict__ starts,
    const int* __restrict__ counts, float* __restrict__ kpart, int Hn, int Dh,
    int ldq) {
  const int bh = blockIdx.x;
  const int b = bh / Hn, h = bh % Hn;
  const int start = starts[b], count = counts[b];
  const int nsplit = gridDim.y;
  const int chunk = (count + nsplit - 1) / nsplit;
  const int kb = blockIdx.y * chunk;
  const int ke = (kb + chunk < count) ? kb + chunk : count;
  const int m = threadIdx.x;
  float s = 0.0f;
  for (int k = kb; k < ke; ++k)
    s += Ks[(size_t)(start + k) * ldq + (size_t)h * Dh + m];
  kpart[((size_t)blockIdx.y * gridDim.x + bh) * Dh + m] = s;
}

__global__ void tc_ksum_reduce(const float* __restrict__ kpart,
                               float* __restrict__ ksum, int nbh, int Dh,
                               int nsplit) {
  const int idx = blockIdx.x * blockDim.x + threadIdx.x;
  if (idx < nbh * Dh) {
    float s = 0.0f;
    for (int j = 0; j < nsplit; ++j) s += kpart[(size_t)j * nbh * Dh + idx];
    ksum[idx] = s;
  }
}

// ---------------------------------------------------------------------------
// Epilogue per node: denom, mean over heads, residual, LN, ReLU
// ---------------------------------------------------------------------------
__global__ __launch_bounds__(256) void tc_epilogue(
    const float* __restrict__ Qs, const float* __restrict__ Vs,
    const float* __restrict__ Numo, const float* __restrict__ ksum,
    const float* __restrict__ scal, const int* __restrict__ batch,
    const int* __restrict__ counts, const float* __restrict__ hin,
    const float* __restrict__ lnw, const float* __restrict__ lnb,
    float* __restrict__ hout, int Hn, int Dh, int ldq) {
  __shared__ float red[256];
  const int n = blockIdx.x;
  const int t = threadIdx.x;
  const int b = batch[n];
  const float inv = scal[2];
  const float Nb = (float)counts[b];
  float val = 0.0f;
  for (int hh = 0; hh < Hn; ++hh) {
    const size_t off = (size_t)n * ldq + (size_t)hh * Dh + t;
    const float q = Qs[off];
    const float kc = ksum[(size_t)(b * Hn + hh) * Dh + t];
    const float dsum = block_sum_256(q * kc, red);
    const float den = inv * dsum + Nb;
    const float numv = inv * Numo[off] + Nb * Vs[off];
    val += numv / den;
  }
  val *= (1.0f / 4.0f);  // mean over H=4 heads
  const float hc = 0.5f * (val + hin[(size_t)n * Dh + t]);
  const float mean = block_sum_256(hc, red) * (1.0f / 256.0f);
  const float d = hc - mean;
  const float var = block_sum_256(d * d, red) * (1.0f / 256.0f);
  const float y = d * rsqrtf(var + 1e-5f) * lnw[t] + lnb[t];
  hout[(size_t)n * Dh + t] = fmaxf(y, 0.0f);
}

// ---------------------------------------------------------------------------
// Orchestration
// ---------------------------------------------------------------------------
extern "C" void kernel_launch(void* const* d_in, const int* in_sizes, int n_in,
                              void* d_out, int out_size, void* d_ws,
                              size_t ws_size, hipStream_t stream) {
  (void)n_in; (void)out_size; (void)ws_size;
  const float* x = (const float*)d_in[0];
  const int* batch = (const int*)d_in[1];
  const float* fc0_w = (const float*)d_in[2];
  const float* fc0_b = (const float*)d_in[3];
  const float* ln_w[3] = {(const float*)d_in[4], (const float*)d_in[6],
                          (const float*)d_in[8]};
  const float* ln_b[3] = {(const float*)d_in[5], (const float*)d_in[7],
                          (const float*)d_in[9]};

  const int Cin = 16, Dh = 256, Hn = 4, ldq = 1024;
  const int N = in_sizes[0] / Cin;  // 32768
  const int B = 16;  // num_graphs (harness-fixed; value only exists on device)
  const int KSPLIT = 16;
  const int tilesPerGraph = ((N / B) + BM - 1) / BM;

  float* ws = (float*)d_ws;
  size_t off = 0;
  float* hbuf = ws + off;  off += (size_t)N * Dh;        //  32 MB
  float* qsb = ws + off;   off += (size_t)N * ldq;       // 128 MB
  float* ksb = ws + off;   off += (size_t)N * ldq;       // 128 MB (reused as Numo)
  float* vsb = ws + off;   off += (size_t)N * ldq;       // 128 MB
  float* kvsb = ws + off;  off += (size_t)B * Hn * Dh * Dh;   // 16 MB
  float* ksumb = ws + off; off += (size_t)B * Hn * Dh;
  float* kpart = ws + off; off += (size_t)KSPLIT * B * Hn * Dh;
  float* part = ws + off;  off += 1024;
  float* scal = ws + off;  off += 8;
  int* counts = (int*)(ws + off);  off += B;
  int* startsb = (int*)(ws + off); off += B;

  // graph structure (recomputed every call; integer atomics -> deterministic)
  tc_zero_i<<<1, 64, 0, stream>>>(counts, B);
  tc_count<<<(N + 255) / 256, 256, 0, stream>>>(batch, counts, N);
  tc_starts<<<1, 1, 0, stream>>>(counts, startsb, B);

  tc_fc0_ln_relu<<<N, 256, 0, stream>>>(x, fc0_w, fc0_b, ln_w[0], ln_b[0], hbuf,
                                        Cin, Dh);

  for (int layer = 0; layer < 2; ++layer) {
    const float* wq = (const float*)d_in[10 + 6 * layer];
    const float* bq = (const float*)d_in[11 + 6 * layer];
    const float* wk = (const float*)d_in[12 + 6 * layer];
    const float* bk = (const float*)d_in[13 + 6 * layer];
    const float* wv = (const float*)d_in[14 + 6 * layer];
    const float* bv = (const float*)d_in[15 + 6 * layer];

    const dim3 g1(ldq / BN, N / BM);
    tc_gemm_hw<<<g1, 128, 0, stream>>>(hbuf, wq, bq, qsb, N, Dh, ldq);
    tc_gemm_hw<<<g1, 128, 0, stream>>>(hbuf, wk, bk, ksb, N, Dh, ldq);
    tc_gemm_hw<<<g1, 128, 0, stream>>>(hbuf, wv, bv, vsb, N, Dh, ldq);

    tc_sumsq_partial<<<1024, 256, 0, stream>>>(qsb, (size_t)N * ldq, part);
    tc_sumsq_final<<<1, 256, 0, stream>>>(part, scal + 0);
    tc_sumsq_partial<<<1024, 256, 0, stream>>>(ksb, (size_t)N * ldq, part);
    tc_sumsq_final<<<1, 256, 0, stream>>>(part, scal + 1);
    tc_inv<<<1, 1, 0, stream>>>(scal);

    tc_ksum_part<<<dim3(B * Hn, KSPLIT), Dh, 0, stream>>>(ksb, startsb, counts,
                                                          kpart, Hn, Dh, ldq);
    tc_ksum_reduce<<<(B * Hn * Dh + 255) / 256, 256, 0, stream>>>(
        kpart, ksumb, B * Hn, Dh, KSPLIT);

    tc_gemm_kvs<<<dim3(Dh / BN, Dh / BM, B * Hn), 128, 0, stream>>>(
        ksb, vsb, startsb, counts, kvsb, Hn, Dh, ldq);
    // ksb is dead after kvs/ksum/sumsq -> reuse it as the num output
    tc_gemm_num<<<dim3(Dh / BN, B * tilesPerGraph, Hn), 128, 0, stream>>>(
        qsb, kvsb, startsb, counts, ksb, Hn, Dh, ldq, tilesPerGraph);

    float* outp = (layer == 1) ? (float*)d_out : hbuf;
    tc_epilogue<<<N, 256, 0, stream>>>(qsb, vsb, ksb, ksumb, scal, batch,
                                       counts, hbuf, ln_w[layer + 1],
                                       ln_b[layer + 1], outp, Hn, Dh, ldq);
  }
}